// StyleConv3d_34935263986182
// MI455X (gfx1250) — compile-verified
//
#include <hip/hip_runtime.h>
#include <hip/hip_bf16.h>
#include <math.h>

// ---------------------------------------------------------------------------
// StyleConv3d for MI455X (gfx1250), fp32 end-to-end (memory-bound; f32 WMMA
// compute is ~100x below the 23.3 TB/s memory floor). Two kernels:
//   1) style MLP + modulate/demodulate -> WMMA-fragment weight layout in d_ws
//   2) implicit-GEMM 3D conv: TDM (tensor_load_to_lds) stages the input patch,
//      432-deep V_WMMA_F32_16X16X4_F32 chain per wave computes 16x16 tiles.
// ---------------------------------------------------------------------------

typedef __attribute__((ext_vector_type(2))) float        v2f;
typedef __attribute__((ext_vector_type(8))) float        v8f;
typedef __attribute__((ext_vector_type(4))) unsigned int u32x4;
typedef __attribute__((ext_vector_type(8))) int          i32x8;
typedef __attribute__((ext_vector_type(4))) int          i32x4;

#define N_      4
#define CIN_    64
#define COUT_   64
#define D_      64
#define DOUT_   62
#define STYLE_  128
#define HID_    128
#define KVOL_   27                      // 3*3*3
#define KTOT_   (CIN_ * KVOL_)          // 1728
#define WAPER_  (KTOT_ * COUT_)         // 110592 floats per sample

// LDS tile: [cin=64][z=3][y=3][x=20]; x padded 18->20 (tile_dim0 % 4B == 0,
// and rows start 16B-aligned since x0 is a multiple of 16).
#define XPAD_     20
#define TILE_LDS  (CIN_ * 3 * 3 * XPAD_)   // 11520 floats = 46080 B

// ---------------------------------------------------------------------------
// Kernel 1: style MLP -> style[n][cin]; modulate + demodulate; emit
// wA[n][chunk][khalf][cout][2] with K = chunk*4 + khalf*2 + j, K = kkk*64+cin
// (exactly the per-lane A-fragment order of V_WMMA_F32_16X16X4_F32).
// ---------------------------------------------------------------------------
__global__ __launch_bounds__(256) void style_mod_kernel(
    const float* __restrict__ s,  const float* __restrict__ w1, const float* __restrict__ b1,
    const float* __restrict__ w2, const float* __restrict__ b2,
    const float* __restrict__ w3, const float* __restrict__ b3,
    const float* __restrict__ weight, float* __restrict__ wA)
{
    __shared__ float h1[HID_], h2[HID_], st[CIN_], fac[COUT_], red[256];
    const int n = blockIdx.x;
    const int t = threadIdx.x;

    if (t < HID_) {
        float acc = b1[t];
        for (int k = 0; k < STYLE_; ++k) acc += s[n * STYLE_ + k] * w1[t * STYLE_ + k];
        h1[t] = acc > 0.f ? acc : 0.01f * acc;
    }
    __syncthreads();
    if (t < HID_) {
        float acc = b2[t];
        for (int k = 0; k < HID_; ++k) acc += h1[k] * w2[t * HID_ + k];
        h2[t] = acc > 0.f ? acc : 0.01f * acc;
    }
    __syncthreads();
    if (t < CIN_) {
        float acc = b3[t];
        for (int k = 0; k < HID_; ++k) acc += h2[k] * w3[t * HID_ + k];
        st[t] = acc;
    }
    __syncthreads();

    // demod factor per cout: rsqrt(sum_{cin,k} (w*style)^2 + eps)
    {
        const int cout = t >> 2, part = t & 3;
        float sum = 0.f;
        for (int i = part; i < KTOT_; i += 4) {
            const int cin = i / KVOL_;
            const float w = weight[cout * KTOT_ + i] * st[cin];
            sum += w * w;
        }
        red[t] = sum;
        __syncthreads();
        if (part == 0)
            fac[cout] = rsqrtf(red[t] + red[t + 1] + red[t + 2] + red[t + 3] + 1e-8f);
    }
    __syncthreads();

    for (int o = t; o < WAPER_; o += 256) {
        const int j     = o & 1;
        const int cout  = (o >> 1) & 63;
        const int khalf = (o >> 7) & 1;
        const int chunk = o >> 8;
        const int K     = chunk * 4 + khalf * 2 + j;   // K = kkk*64 + cin
        const int kkk   = K >> 6;
        const int cin   = K & 63;
        wA[n * WAPER_ + o] = weight[(cout * CIN_ + cin) * KVOL_ + kkk] * st[cin] * fac[cout];
    }
}

// ---------------------------------------------------------------------------
// Kernel 2: implicit GEMM conv.
// Grid: (n*4 + xtile, zo, yo); block = 128 (4 waves).
// TDM stages the 64x3x3x20 input patch into LDS (wave 0 issues descriptor,
// s_wait_tensorcnt 0, block barrier). Each wave then runs the 432-step f32
// WMMA chain for its 16(cout) x 16(x) tile.
// ---------------------------------------------------------------------------
__global__ __launch_bounds__(128) void styleconv_wmma_kernel(
    const float* __restrict__ x, const float* __restrict__ wA, float* __restrict__ y)
{
    __shared__ float xt[TILE_LDS];

    const int bx  = blockIdx.x;
    const int n   = bx >> 2;
    const int x0  = (bx & 3) * 16;
    const int zo  = blockIdx.y;
    const int yo  = blockIdx.z;
    const int tid = threadIdx.x;

    // ---- TDM: DMA the input patch into LDS (one descriptor per block) ----
    if (tid < 32) {
        const unsigned long long gaddr =
            (unsigned long long)(uintptr_t)x +
            4ull * (((((unsigned long long)n * CIN_) * D_ + zo) * D_ + yo) * D_ + x0);
        const unsigned lds_base = (unsigned)(uintptr_t)xt;   // low 32 bits = LDS offset

        // D# group 0: count=1 | lds_addr | global_addr(57b) | type=2
        u32x4 g0;
        g0[0] = 1u;
        g0[1] = lds_base;
        g0[2] = (unsigned)(gaddr & 0xffffffffull);
        g0[3] = (unsigned)((gaddr >> 32) & 0x1ffffffull) | (2u << 30);

        // D# group 1: data_size=4B; tensor_dim0 = 64-x0 (x OOB -> zero fill,
        // dims are relative to the tile-start address); tensor_dim1 = 64;
        // tile = 20 x 3 x 3; strides: dim0->y = 64, dim1->z = 4096 elements.
        i32x8 g1;
        g1[0] = (int)(2u << 16);                       // data_size = 2 (4 bytes)
        g1[1] = (int)((unsigned)(D_ - x0) << 16);      // tensor_dim0[15:0]
        g1[2] = (int)(64u << 16);                      // tensor_dim1[15:0]
        g1[3] = (int)(20u << 16);                      // tile_dim0 = 20
        g1[4] = (int)((3u << 16) | 3u);                // tile_dim2 | tile_dim1
        g1[5] = (int)64;                               // tensor_dim0_stride lo
        g1[6] = (int)(4096u << 16);                    // tensor_dim1_stride lo16
        g1[7] = 0;                                     // tensor_dim1_stride hi

        // D# group 2: tensor_dim2 = 64 (z), tensor_dim3 = 64 (cin),
        // tensor_dim2_stride = 262144 (z->cin = D^3), tile_dim3 = 64.
        i32x4 g2;
        g2[0] = 64;
        g2[1] = 64;
        g2[2] = 262144;
        g2[3] = (int)(64u << 16);

        // D# group 3: no 5th dim.
        i32x4 g3;
        g3[0] = 0; g3[1] = 0; g3[2] = 0; g3[3] = 0;

        // Extra 5th operand on this toolchain (clang-23 6-arg form): zero-fill.
        i32x8 gz;
        gz[0] = 0; gz[1] = 0; gz[2] = 0; gz[3] = 0;
        gz[4] = 0; gz[5] = 0; gz[6] = 0; gz[7] = 0;

        __builtin_amdgcn_tensor_load_to_lds(g0, g1, g2, g3, gz, 0);
        __builtin_amdgcn_s_wait_tensorcnt(0);
    }
    __syncthreads();

    const int wave  = tid >> 5;
    const int lane  = tid & 31;
    const int xl    = lane & 15;    // B/C column (output x), also A's M
    const int khalf = lane >> 4;    // which K pair this lane holds
    const int coutA = wave * 16 + xl;

    // Per-lane hoisted base pointers: A index = chunk*256 + khalf*128 + coutA*2
    const float* Ap = wA + n * WAPER_ + khalf * 128 + coutA * 2;

    v8f c = {};
    for (int kkk = 0; kkk < KVOL_; ++kkk) {
        const int kz  = kkk / 9;
        const int kyr = kkk - kz * 9;
        const int ky  = kyr / 3;
        const int kx  = kyr - ky * 3;
        // B base: cin0 = cc*4 + khalf*2; element [cin0][kz][ky][xl+kx]
        const float* Bp = xt + khalf * 2 * (3 * 3 * XPAD_) +
                          (kz * 3 + ky) * XPAD_ + xl + kx;
        const float* Ak = Ap + kkk * 4096;
        #pragma unroll 4
        for (int cc = 0; cc < 16; ++cc) {
            const v2f a = *(const v2f*)(Ak + cc * 256);
            v2f b;
            b.x = Bp[cc * (4 * 3 * 3 * XPAD_)];
            b.y = Bp[cc * (4 * 3 * 3 * XPAD_) + (3 * 3 * XPAD_)];
            c = __builtin_amdgcn_wmma_f32_16x16x4_f32(
                    /*neg_a=*/false, a, /*neg_b=*/false, b,
                    /*c_mod=*/(short)0, c, /*reuse_a=*/false, /*reuse_b=*/false);
        }
    }

    // ---- store: C/D layout lane<16 -> M=r, lane>=16 -> M=r+8; N = lane%16 ----
    const int gxo = x0 + xl;
    if (gxo < DOUT_) {
        #pragma unroll
        for (int r = 0; r < 8; ++r) {
            const int co = wave * 16 + khalf * 8 + r;
            y[(((n * COUT_ + co) * DOUT_ + zo) * DOUT_ + yo) * DOUT_ + gxo] = c[r];
        }
    }
}

// ---------------------------------------------------------------------------
extern "C" void kernel_launch(void* const* d_in, const int* in_sizes, int n_in,
                              void* d_out, int out_size, void* d_ws, size_t ws_size,
                              hipStream_t stream)
{
    const float* x      = (const float*)d_in[0];
    const float* s      = (const float*)d_in[1];
    const float* w1     = (const float*)d_in[2];
    const float* b1     = (const float*)d_in[3];
    const float* w2     = (const float*)d_in[4];
    const float* b2     = (const float*)d_in[5];
    const float* w3     = (const float*)d_in[6];
    const float* b3     = (const float*)d_in[7];
    const float* weight = (const float*)d_in[8];
    float*       y      = (float*)d_out;
    float*       wA     = (float*)d_ws;   // N * 110592 floats = 1.77 MB

    style_mod_kernel<<<N_, 256, 0, stream>>>(s, w1, b1, w2, b2, w3, b3, weight, wA);

    dim3 grid(N_ * 4, DOUT_, DOUT_);      // (n, x-tile) x zo x yo
    styleconv_wmma_kernel<<<grid, 128, 0, stream>>>(x, wA, y);
}